// StructNConv2D_s_with_d_22660247454232
// MI455X (gfx1250) — compile-verified
//
#include <hip/hip_runtime.h>
#include <math.h>

#define EPSV 1e-20f

typedef __attribute__((ext_vector_type(2))) float v2f;
typedef __attribute__((ext_vector_type(8))) float v8f;

__device__ __forceinline__ float softplus_f(float x) {
    // log(1 + exp(x)), stable
    if (x > 20.0f) return x;
    return log1pf(expf(x));
}

// ---------------------------------------------------------------------------
// k0: tiny weight preprocessing (softplus + reductions). One thread; trivial.
// scal[0]=sum(softplus(spatial_weight)) scal[1]=sum(softplus(channel_weight))
// scal[2]=softplus(w_prop)
// ---------------------------------------------------------------------------
__global__ void k0_weights(const float* __restrict__ sw,
                           const float* __restrict__ cww,
                           const float* __restrict__ wprop,
                           float* __restrict__ swS,
                           float* __restrict__ cwS,
                           float* __restrict__ scal) {
    float ssum = 0.f;
    for (int i = 0; i < 64 * 9; ++i) { float v = softplus_f(sw[i]); swS[i] = v; ssum += v; }
    float csum = 0.f;
    for (int i = 0; i < 64 * 64; ++i) { float v = softplus_f(cww[i]); cwS[i] = v; csum += v; }
    scal[0] = ssum;
    scal[1] = csum;
    scal[2] = softplus_f(wprop[0]);
}

// ---------------------------------------------------------------------------
// k1: fused 3x3 argmax-select (two score functions over one LDS halo tile)
//     + elementwise s_prop / cs_prop chain.
//     Writes t0 = cs_prop*s_prop, t1 = cs_prop.
// ---------------------------------------------------------------------------
__global__ __launch_bounds__(256)
void k1_prop(const float* __restrict__ d, const float* __restrict__ cd,
             const float* __restrict__ s, const float* __restrict__ cs,
             const float* __restrict__ wsd, const float* __restrict__ scal,
             float* __restrict__ t0, float* __restrict__ t1,
             int H, int W) {
    __shared__ float sd[18 * 18];
    __shared__ float sc[18 * 18];

    const int tilesX = W >> 4;
    const int plane  = blockIdx.y;                 // b*C + c
    const int x0 = (blockIdx.x % tilesX) << 4;
    const int y0 = (blockIdx.x / tilesX) << 4;
    const int planeOff = plane * H * W;
    const int tid = threadIdx.x;

    for (int i = tid; i < 18 * 18; i += 256) {
        int ly = i / 18, lx = i % 18;
        int gy = y0 + ly - 1, gx = x0 + lx - 1;
        float dv = 0.f, cv = 0.f;
        if (gy >= 0 && gy < H && gx >= 0 && gx < W) {
            int idx = planeOff + gy * W + gx;
            dv = d[idx];
            cv = cd[idx];
        }
        sd[i] = dv;
        sc[i] = cv;
    }
    __syncthreads();

    const int lx = tid & 15, ly = tid >> 4;
    // both scores >= 0 in-bounds; OOB encodes (0,0) -> score 0 never wins strict '>'
    float bs1 = -1.f, dmax = 0.f, cdmax = 0.f;
    float bs2 = -1.f, dmin = 0.f, cdmin = 0.f;
#pragma unroll
    for (int di = 0; di < 3; ++di)
#pragma unroll
        for (int dj = 0; dj < 3; ++dj) {
            int li = (ly + di) * 18 + (lx + dj);
            float dv = sd[li], cv = sc[li];
            float s1 = dv * cv;
            float s2 = cv / (dv + EPSV);
            if (s1 > bs1) { bs1 = s1; dmax = dv; cdmax = cv; }
            if (s2 > bs2) { bs2 = s2; dmin = dv; cdmin = cv; }
        }

    const int idx = planeOff + (y0 + ly) * W + (x0 + lx);
    const float sv  = s[idx];
    const float csv = cs[idx];
    const float w0 = wsd[0], w1 = wsd[1], wp = scal[2];

    float m   = fabsf(dmin / (dmax + EPSV));
    float sfd = (1.f - w0 - w1) * m + w0 * m * m + w1 * m * m * m;
    float cfd = cdmax * cdmin;
    float num = wp * csv * sv + cfd * sfd;
    float den = wp * csv + cfd;
    float s_prop  = num / (den + EPSV);
    float cs_prop = den / (wp + 1.f);

    t0[idx] = cs_prop * s_prop;
    t1[idx] = cs_prop;
}

// ---------------------------------------------------------------------------
// k2: depthwise 3x3 conv (correlation, zero pad) on t0 (nom) and t1 (den).
//     Writes t2 = cs_spatial*s_spatial, t3 = cs_spatial into d_out (scratch).
// ---------------------------------------------------------------------------
__global__ __launch_bounds__(256)
void k2_spatial(const float* __restrict__ t0, const float* __restrict__ t1,
                const float* __restrict__ swS, const float* __restrict__ scal,
                float* __restrict__ outNom, float* __restrict__ outDen,
                int H, int W, int C) {
    __shared__ float a0[18 * 18];
    __shared__ float a1[18 * 18];
    __shared__ float wloc[9];

    const int tilesX = W >> 4;
    const int plane  = blockIdx.y;
    const int c      = plane & (C - 1);
    const int x0 = (blockIdx.x % tilesX) << 4;
    const int y0 = (blockIdx.x / tilesX) << 4;
    const int planeOff = plane * H * W;
    const int tid = threadIdx.x;

    if (tid < 9) wloc[tid] = swS[c * 9 + tid];
    for (int i = tid; i < 18 * 18; i += 256) {
        int ly = i / 18, lx = i % 18;
        int gy = y0 + ly - 1, gx = x0 + lx - 1;
        float v0 = 0.f, v1 = 0.f;
        if (gy >= 0 && gy < H && gx >= 0 && gx < W) {
            int idx = planeOff + gy * W + gx;
            v0 = t0[idx];
            v1 = t1[idx];
        }
        a0[i] = v0;
        a1[i] = v1;
    }
    __syncthreads();

    const int lx = tid & 15, ly = tid >> 4;
    float nom = 0.f, den = 0.f;
#pragma unroll
    for (int di = 0; di < 3; ++di)
#pragma unroll
        for (int dj = 0; dj < 3; ++dj) {
            float w = wloc[di * 3 + dj];
            int li = (ly + di) * 18 + (lx + dj);
            nom += a0[li] * w;
            den += a1[li] * w;
        }

    const float invSumSW = 1.f / scal[0];
    float s_sp  = nom / (den + EPSV);
    float cs_sp = den * invSumSW;

    const int idx = planeOff + (y0 + ly) * W + (x0 + lx);
    outNom[idx] = cs_sp * s_sp;
    outDen[idx] = cs_sp;
}

// ---------------------------------------------------------------------------
// k3: channel-mixing GEMM via V_WMMA_F32_16X16X4_F32.
//     out[0:N)   holds t2 on entry, s_out on exit
//     out[N:2N)  holds t3 on entry, cs_out on exit
//     Each wave: 16 pixel columns x full M=64 (4 M-tiles) x 2 accumulators,
//     K=64 fully unrolled -> 128 WMMA per wave. Wave-uniform address math is
//     scalarized via readfirstlane so B loads use SGPR-base + 32-bit voffset
//     (GVS) with immediate channel-stride offsets.  Safe in-place: each pixel
//     column is owned by exactly one wave (reads all c, then writes all o).
// ---------------------------------------------------------------------------
__global__ __launch_bounds__(256)
void k3_channel(float* __restrict__ out,
                const float* __restrict__ cwS,
                const float* __restrict__ scal,
                const float* __restrict__ bias) {
    constexpr int HW = 320 * 320;              // 102400
    constexpr int N  = 8 * 64 * HW;            // 52,428,800

    __shared__ float lcw[64 * 64];             // softplus(channel_weight) [o][c]
    __shared__ float lbias[64];
    const int tid = threadIdx.x;
    for (int i = tid; i < 64 * 64; i += 256) lcw[i] = cwS[i];
    if (tid < 64) lbias[tid] = bias[tid];
    __syncthreads();

    const int lane = tid & 31;
    // wave-uniform scalars (force into SGPRs)
    const int wl  = __builtin_amdgcn_readfirstlane((int)(blockIdx.x * 8) + (tid >> 5));
    const int p0  = wl << 4;                   // 16 pixels per wave
    const int b   = p0 / HW;
    const int hw0 = p0 % HW;

    const int n    = lane & 15;                // pixel within tile / N index
    const int koff = (lane >> 4) << 1;         // 0 for lanes 0-15, 2 for 16-31
    const int mrow = lane & 15;                // A-matrix row within tile

    const float* __restrict__ X2 = out + b * 64 * HW + hw0;   // scalar base
    const float* __restrict__ X3 = X2 + N;
    const int vb = koff * HW + n;              // per-lane element offset (c=koff)

    v8f accN[4] = {v8f{}, v8f{}, v8f{}, v8f{}};
    v8f accD[4] = {v8f{}, v8f{}, v8f{}, v8f{}};

#pragma unroll
    for (int kb = 0; kb < 16; ++kb) {
        const int o0 = vb + kb * 4 * HW;
        v2f bN, bD;
        bN[0] = X2[o0];
        bN[1] = X2[o0 + HW];
        bD[0] = X3[o0];
        bD[1] = X3[o0 + HW];
        const int arow = mrow * 64 + (kb << 2) + koff;
#pragma unroll
        for (int t = 0; t < 4; ++t) {
            v2f a;
            a[0] = lcw[(t << 10) + arow];
            a[1] = lcw[(t << 10) + arow + 1];
            accN[t] = __builtin_amdgcn_wmma_f32_16x16x4_f32(
                false, a, false, bN, (short)0, accN[t], false, false);
            accD[t] = __builtin_amdgcn_wmma_f32_16x16x4_f32(
                false, a, false, bD, (short)0, accD[t], false, false);
        }
    }

    const float invSumCW = 1.f / scal[1];
    const int mhalf = (lane >> 4) << 3;        // 0 or 8 (C/D layout lane-half)
    float* __restrict__ Sout  = out + b * 64 * HW + hw0;   // overwrite t2
    float* __restrict__ CSout = Sout + N;                  // overwrite t3

#pragma unroll
    for (int t = 0; t < 4; ++t) {
        const int obase = (t << 4) + mhalf;
        const int vo = obase * HW + n;
#pragma unroll
        for (int r = 0; r < 8; ++r) {
            const float nv = accN[t][r];
            const float dv = accD[t][r];
            Sout[vo + r * HW]  = nv / (dv + EPSV) + lbias[obase + r];
            CSout[vo + r * HW] = dv * invSumCW;
        }
    }
}

// ---------------------------------------------------------------------------
extern "C" void kernel_launch(void* const* d_in, const int* in_sizes, int n_in,
                              void* d_out, int out_size, void* d_ws, size_t ws_size,
                              hipStream_t stream) {
    const float* d     = (const float*)d_in[0];
    const float* cd    = (const float*)d_in[1];
    const float* s     = (const float*)d_in[2];
    const float* cs    = (const float*)d_in[3];
    const float* wsd   = (const float*)d_in[4];   // w_s_from_d (2)
    const float* wprop = (const float*)d_in[5];   // w_prop (1)
    const float* sw    = (const float*)d_in[6];   // spatial_weight (64*9)
    const float* cww   = (const float*)d_in[7];   // channel_weight (64*64)
    const float* bias  = (const float*)d_in[8];   // (64)

    float* out = (float*)d_out;
    float* ws  = (float*)d_ws;

    const int B = 8, C = 64, H = 320, W = 320;
    const int HW = H * W;
    const int N  = B * C * HW;                   // 52,428,800 < 2^31

    float* t0   = ws;                // cs_prop * s_prop
    float* t1   = ws + (size_t)N;    // cs_prop
    float* swS  = ws + 2 * (size_t)N;
    float* cwS  = swS + 576;
    float* scal = cwS + 4096;

    hipLaunchKernelGGL(k0_weights, dim3(1), dim3(1), 0, stream,
                       sw, cww, wprop, swS, cwS, scal);

    dim3 gTile((W / 16) * (H / 16), B * C);      // (400, 512)
    hipLaunchKernelGGL(k1_prop, gTile, dim3(256), 0, stream,
                       d, cd, s, cs, wsd, scal, t0, t1, H, W);

    hipLaunchKernelGGL(k2_spatial, gTile, dim3(256), 0, stream,
                       t0, t1, swS, scal, out, out + (size_t)N, H, W, C);

    const int nPix   = B * HW;                   // 819,200
    const int blocks = nPix / (8 * 16);          // 6400 (8 waves/block, 16 px/wave)
    hipLaunchKernelGGL(k3_channel, dim3(blocks), dim3(256), 0, stream,
                       out, cwS, scal, bias);
}